// CGNN_81363860455904
// MI455X (gfx1250) — compile-verified
//
#include <hip/hip_runtime.h>

typedef float v2f __attribute__((ext_vector_type(2)));
typedef float v8f __attribute__((ext_vector_type(8)));

#define D_S        32
#define H_AGG      128
#define C_MID      66
#define CPAD       68      // K padded to multiple of 4
#define B_SZ       16
#define T_SZ       4
#define F_SZ       512
#define S_SZ       14
#define PIX_PER_IMG (F_SZ * S_SZ)               // 7168
#define N_ROWS     (B_SZ * T_SZ * PIX_PER_IMG)  // 458752

// ---- CDNA5 f32 WMMA: D(16x16) = A(16x4) * B(4x16) + C, fp32 exact ----
static __device__ __forceinline__ v8f wmma_f32(v2f a, v2f b, v8f c) {
  return __builtin_amdgcn_wmma_f32_16x16x4_f32(
      /*neg_a=*/false, a, /*neg_b=*/false, b,
      /*c_mod=*/(short)0, c, /*reuse_a=*/false, /*reuse_b=*/false);
}

// =====================================================================
// Kernel 1: aggregate MLP  y = relu(s @ W1 + b1) @ W2 + b2   per row
// 128 threads = 4 waves; each wave owns one 16-row M-tile.
// =====================================================================
__global__ __launch_bounds__(128) void k_agg_gemm(
    const float* __restrict__ s_in,
    const float* __restrict__ w1, const float* __restrict__ b1,
    const float* __restrict__ w2, const float* __restrict__ b2,
    float* __restrict__ y_out)
{
  __shared__ float h_lds[4 * 16 * H_AGG];   // 32 KB
  const int lane = threadIdx.x & 31;
  const int wave = threadIdx.x >> 5;
  const int lm = lane & 15;                 // M (A) / N (B,C)
  const int kh = lane >> 4;                 // K half
  const int k0 = kh * 2;
  const long rowBase = (long)blockIdx.x * 64 + wave * 16;

  // A fragments for GEMM1: X[16 x 32]
  v2f afr[8];
  {
    const float* xrow = s_in + (rowBase + lm) * D_S;
#pragma unroll
    for (int ks = 0; ks < 8; ++ks) {
      int k = 4 * ks + k0;
      afr[ks].x = xrow[k];
      afr[ks].y = xrow[k + 1];
    }
  }

  float* hme = h_lds + wave * 16 * H_AGG;
  // GEMM1: H[16x128] = X * W1, relu(+b1) -> LDS
#pragma unroll
  for (int nt = 0; nt < 8; ++nt) {
    v8f acc = {};
    const int n = nt * 16 + lm;
#pragma unroll
    for (int ks = 0; ks < 8; ++ks) {
      int k = 4 * ks + k0;
      v2f bf;
      bf.x = w1[(size_t)k * H_AGG + n];
      bf.y = w1[(size_t)(k + 1) * H_AGG + n];
      acc = wmma_f32(afr[ks], bf, acc);
    }
    const float bias = b1[n];
#pragma unroll
    for (int v = 0; v < 8; ++v) {
      int m = v + 8 * kh;
      float val = acc[v] + bias;
      hme[m * H_AGG + n] = val > 0.f ? val : 0.f;
    }
  }
  __syncthreads();

  // GEMM2: Y[16x32] = H * W2 (+b2)
#pragma unroll
  for (int nt = 0; nt < 2; ++nt) {
    v8f acc = {};
    const int n = nt * 16 + lm;
#pragma unroll 8
    for (int ks = 0; ks < 32; ++ks) {
      int k = 4 * ks + k0;
      v2f af = *(const v2f*)&hme[lm * H_AGG + k];   // k even -> 8B aligned
      v2f bf;
      bf.x = w2[(size_t)k * D_S + n];
      bf.y = w2[(size_t)(k + 1) * D_S + n];
      acc = wmma_f32(af, bf, acc);
    }
    const float bias = b2[n];
#pragma unroll
    for (int v = 0; v < 8; ++v) {
      long r = rowBase + v + 8 * kh;
      y_out[r * D_S + n] = acc[v] + bias;
    }
  }
}

// =====================================================================
// Kernel 2: a[b,t] = (sum_t' act*y - act_t*y_t) * p,  p = 1/max(cnt-1,>0)
// =====================================================================
__global__ __launch_bounds__(256) void k_cross_t(
    const float* __restrict__ y, const float* __restrict__ act,
    float* __restrict__ a_out)
{
  const long PER_BT = (long)PIX_PER_IMG * D_S;   // 229376
  long tid = (long)blockIdx.x * 256 + threadIdx.x;
  long b = tid / PER_BT;
  long qd = tid - b * PER_BT;
  if (b >= B_SZ) return;

  float a0 = act[b * 4 + 0], a1 = act[b * 4 + 1];
  float a2 = act[b * 4 + 2], a3 = act[b * 4 + 3];
  float p = (a0 + a1 + a2 + a3) - 1.f;
  if (p < 0.f) p = 0.f;
  float pinv = (p == 0.f) ? 1.f : 1.f / p;

  const float* yb = y + b * 4 * PER_BT;
  float y0 = yb[0 * PER_BT + qd] * a0;
  float y1 = yb[1 * PER_BT + qd] * a1;
  float y2 = yb[2 * PER_BT + qd] * a2;
  float y3 = yb[3 * PER_BT + qd] * a3;
  float sum = y0 + y1 + y2 + y3;

  float* ab = a_out + b * 4 * PER_BT;
  ab[0 * PER_BT + qd] = (sum - y0) * pinv;
  ab[1 * PER_BT + qd] = (sum - y1) * pinv;
  ab[2 * PER_BT + qd] = (sum - y2) * pinv;
  ab[3 * PER_BT + qd] = (sum - y3) * pinv;
}

// =====================================================================
// Kernel 3a: v = relu( pw1( dw1( concat(a, s, pe) ) ) )
// Block = 64 pixels (never crosses an image: 7168 % 64 == 0).
// Phase 1: depthwise 3x3 + bias -> a 64-pixel x 68-channel tile in LDS (K-padded with zeros)
// Phase 2: WMMA GEMM K=68, N padded to 80, write cols < 66.
// =====================================================================
__global__ __launch_bounds__(128) void k_mid(
    const float* __restrict__ a_buf, const float* __restrict__ s_in,
    const float* __restrict__ pe,
    const float* __restrict__ dw_w, const float* __restrict__ dw_b,
    const float* __restrict__ pw_w, const float* __restrict__ pw_b,
    float* __restrict__ v_out)
{
  __shared__ float u_lds[64 * CPAD];   // ~17 KB
  const long pixBase = (long)blockIdx.x * 64;

  for (int idx = threadIdx.x; idx < 64 * CPAD; idx += 128) {
    int p = idx / CPAD;
    int c = idx - p * CPAD;
    float val = 0.f;
    if (c < C_MID) {
      long pix = pixBase + p;
      int bt = (int)(pix / PIX_PER_IMG);
      int q  = (int)(pix - (long)bt * PIX_PER_IMG);
      int h = q / S_SZ;
      int w = q - h * S_SZ;
      int t = bt & 3;
      val = dw_b[c];
      const float* wp = dw_w + c * 9;
#pragma unroll
      for (int dy = 0; dy < 3; ++dy) {
        int hh = h + dy - 1;
        if (hh < 0 || hh >= F_SZ) continue;
#pragma unroll
        for (int dx = 0; dx < 3; ++dx) {
          int ww = w + dx - 1;
          if (ww < 0 || ww >= S_SZ) continue;
          long q2 = (long)hh * S_SZ + ww;
          float zv;
          if (c < 32)
            zv = a_buf[((long)bt * PIX_PER_IMG + q2) * D_S + c];
          else if (c < 64)
            zv = s_in[((long)bt * PIX_PER_IMG + q2) * D_S + (c - 32)];
          else
            zv = pe[((long)t * PIX_PER_IMG + q2) * 2 + (c - 64)];
          val += wp[dy * 3 + dx] * zv;
        }
      }
    }
    u_lds[idx] = val;
  }
  __syncthreads();

  const int lane = threadIdx.x & 31;
  const int wave = threadIdx.x >> 5;
  const int lm = lane & 15, kh = lane >> 4, k0 = kh * 2;
  const float* ume = u_lds + wave * 16 * CPAD;

#pragma unroll
  for (int nt = 0; nt < 5; ++nt) {
    v8f acc = {};
    const int n = nt * 16 + lm;
#pragma unroll
    for (int ks = 0; ks < 17; ++ks) {
      int k = 4 * ks + k0;
      v2f af = *(const v2f*)&ume[lm * CPAD + k];
      v2f bf;
      bf.x = (k     < C_MID && n < C_MID) ? pw_w[n * C_MID + k]     : 0.f;
      bf.y = (k + 1 < C_MID && n < C_MID) ? pw_w[n * C_MID + k + 1] : 0.f;
      acc = wmma_f32(af, bf, acc);
    }
    if (n < C_MID) {
      const float bias = pw_b[n];
      long pb = pixBase + wave * 16;
#pragma unroll
      for (int v = 0; v < 8; ++v) {
        long pix = pb + v + 8 * kh;
        float val = acc[v] + bias;
        v_out[pix * C_MID + n] = val > 0.f ? val : 0.f;
      }
    }
  }
}

// =====================================================================
// Kernel 3b: s_out = pw2( dw2(v) ) + s
// =====================================================================
__global__ __launch_bounds__(128) void k_out(
    const float* __restrict__ v_buf, const float* __restrict__ s_in,
    const float* __restrict__ dw_w, const float* __restrict__ dw_b,
    const float* __restrict__ pw_w, const float* __restrict__ pw_b,
    float* __restrict__ s_out)
{
  __shared__ float u_lds[64 * CPAD];
  const long pixBase = (long)blockIdx.x * 64;

  for (int idx = threadIdx.x; idx < 64 * CPAD; idx += 128) {
    int p = idx / CPAD;
    int c = idx - p * CPAD;
    float val = 0.f;
    if (c < C_MID) {
      long pix = pixBase + p;
      int bt = (int)(pix / PIX_PER_IMG);
      int q  = (int)(pix - (long)bt * PIX_PER_IMG);
      int h = q / S_SZ;
      int w = q - h * S_SZ;
      val = dw_b[c];
      const float* wp = dw_w + c * 9;
#pragma unroll
      for (int dy = 0; dy < 3; ++dy) {
        int hh = h + dy - 1;
        if (hh < 0 || hh >= F_SZ) continue;
#pragma unroll
        for (int dx = 0; dx < 3; ++dx) {
          int ww = w + dx - 1;
          if (ww < 0 || ww >= S_SZ) continue;
          long q2 = (long)hh * S_SZ + ww;
          val += wp[dy * 3 + dx] *
                 v_buf[((long)bt * PIX_PER_IMG + q2) * C_MID + c];
        }
      }
    }
    u_lds[idx] = val;
  }
  __syncthreads();

  const int lane = threadIdx.x & 31;
  const int wave = threadIdx.x >> 5;
  const int lm = lane & 15, kh = lane >> 4, k0 = kh * 2;
  const float* ume = u_lds + wave * 16 * CPAD;

#pragma unroll
  for (int nt = 0; nt < 2; ++nt) {
    v8f acc = {};
    const int n = nt * 16 + lm;   // n < 32 always valid
#pragma unroll
    for (int ks = 0; ks < 17; ++ks) {
      int k = 4 * ks + k0;
      v2f af = *(const v2f*)&ume[lm * CPAD + k];
      v2f bf;
      bf.x = (k     < C_MID) ? pw_w[n * C_MID + k]     : 0.f;
      bf.y = (k + 1 < C_MID) ? pw_w[n * C_MID + k + 1] : 0.f;
      acc = wmma_f32(af, bf, acc);
    }
    const float bias = pw_b[n];
    long pb = pixBase + wave * 16;
#pragma unroll
    for (int v = 0; v < 8; ++v) {
      long pix = pb + v + 8 * kh;
      s_out[pix * D_S + n] = acc[v] + bias + s_in[pix * D_S + n];
    }
  }
}

// =====================================================================
extern "C" void kernel_launch(void* const* d_in, const int* in_sizes, int n_in,
                              void* d_out, int out_size, void* d_ws, size_t ws_size,
                              hipStream_t stream)
{
  (void)in_sizes; (void)n_in; (void)out_size; (void)ws_size;
  const float* s0  = (const float*)d_in[0];
  const float* pe  = (const float*)d_in[1];
  const float* act = (const float*)d_in[2];
  const float* aw1 = (const float*)d_in[3];
  const float* ab1 = (const float*)d_in[4];
  const float* aw2 = (const float*)d_in[5];
  const float* ab2 = (const float*)d_in[6];
  const float* hdw = (const float*)d_in[7];
  const float* hdb = (const float*)d_in[8];
  const float* hpw = (const float*)d_in[9];
  const float* hpb = (const float*)d_in[10];
  const float* odw = (const float*)d_in[11];
  const float* odb = (const float*)d_in[12];
  const float* opw = (const float*)d_in[13];
  const float* opb = (const float*)d_in[14];

  float* ws    = (float*)d_ws;
  float* y_buf = ws;
  float* a_buf = y_buf + (size_t)N_ROWS * D_S;
  float* v_buf = a_buf + (size_t)N_ROWS * D_S;
  float* s_tmp = v_buf + (size_t)N_ROWS * C_MID;
  float* out   = (float*)d_out;

  const float* s_cur = s0;
  for (int it = 0; it < 2; ++it) {
    float* s_nxt = (it == 1) ? out : s_tmp;
    k_agg_gemm<<<N_ROWS / 64, 128, 0, stream>>>(s_cur, aw1, ab1, aw2, ab2, y_buf);
    k_cross_t<<<(B_SZ * PIX_PER_IMG * D_S) / 256, 256, 0, stream>>>(y_buf, act, a_buf);
    k_mid<<<N_ROWS / 64, 128, 0, stream>>>(a_buf, s_cur, pe, hdw, hdb, hpw, hpb, v_buf);
    k_out<<<N_ROWS / 64, 128, 0, stream>>>(v_buf, s_cur, odw, odb, opw, opb, s_nxt);
    s_cur = s_nxt;
  }
}